// RBNGr_50766513438852
// MI455X (gfx1250) — compile-verified
//
#include <hip/hip_runtime.h>

// RBNGr (Riemannian batch norm on Grassmann Gr(96,16)) — fused CDNA5 kernel.
// B=64, C=128, N=96, P=16. One workgroup per channel, 4 wave32 per WG.
// All small GEMMs via V_WMMA_F32_16X16X4_F32; SVDs via one-sided Jacobi in LDS;
// pinv via Newton-Schulz (WMMA); X tiles via async global->LDS (gfx1250).

#define B_N 64
#define C_N 128
#define N_N 96
#define P_N 16
#define TILE_ELEMS (N_N * P_N)   // 1536 floats per (b,c) tile
#define LDP 20                   // LDS leading dim (pad 16->20: bank spread, rows 16B aligned)
#define NW 4                     // waves per workgroup
#define NS_ITERS 14              // Newton-Schulz iterations for 16x16 inverse
#define JACOBI_SWEEPS 7          // one-sided Jacobi sweeps

typedef __attribute__((ext_vector_type(2))) float v2f;
typedef __attribute__((ext_vector_type(8))) float v8f;
typedef __attribute__((ext_vector_type(4))) int v4i;

#define AS1 __attribute__((address_space(1)))
#define AS3 __attribute__((address_space(3)))

#if __has_builtin(__builtin_amdgcn_global_load_async_to_lds_b128) && \
    __has_builtin(__builtin_amdgcn_s_wait_asynccnt)
#define RB_ASYNC_LDS 1
#else
#define RB_ASYNC_LDS 0
#endif

__device__ __forceinline__ int laneid() { return (int)(threadIdx.x & 31u); }

__device__ __forceinline__ float wave_sum(float v) {
#pragma unroll
  for (int m = 16; m >= 1; m >>= 1) v += __shfl_xor(v, m, 32);
  return v;
}

// ---- WMMA wrapper: D = A(16x4,f32) * B(4x16,f32) + C ------------------------
__device__ __forceinline__ v8f wmma4(v2f a, v2f b, v8f c) {
  // 8 args: (neg_a, A, neg_b, B, c_mod, C, reuse_a, reuse_b)
  return __builtin_amdgcn_wmma_f32_16x16x4_f32(false, a, false, b, (short)0, c,
                                               false, false);
}

// A-tile fragment layout (16x4 f32): lanes 0-15 -> M=lane, vgpr {K0,K1};
// lanes 16-31 -> M=lane-16, vgpr {K2,K3}.
// B-tile fragment layout (4x16 f32): lanes 0-15 -> N=lane, rows {K0(K2)} etc.
// C/D: vgpr r, lanes 0-15 -> M=r, lanes 16-31 -> M=r+8, N=lane&15.

// D(16x16, stride LDP) = A(96x16)^T * Bm(96x16); A,B,D in LDS.
__device__ __forceinline__ void mm_AtB96(const float* A, const float* Bm, float* D) {
  const int l = laneid(), m = l & 15, kb = (l < 16) ? 0 : 2, mb = (l < 16) ? 0 : 8;
  v8f acc = {};
#pragma unroll
  for (int k0 = 0; k0 < N_N; k0 += 4) {
    v2f a, b;
    a.x = A[(k0 + kb) * LDP + m];      // A^T[m][k] = A[k][m]
    a.y = A[(k0 + kb + 1) * LDP + m];
    b.x = Bm[(k0 + kb) * LDP + m];     // B[k][n], n == l&15
    b.y = Bm[(k0 + kb + 1) * LDP + m];
    acc = wmma4(a, b, acc);
  }
#pragma unroll
  for (int r = 0; r < 8; ++r) D[(mb + r) * LDP + m] = acc[r];
  __builtin_amdgcn_wave_barrier();
}

// D(96x16, ldD) (+)= A(96x16, LDP) * op(B16); op = transB ? B^T : B.
__device__ __forceinline__ void mm_NxP_P(const float* A, const float* B16, float* D,
                                         int ldD, bool transB, bool accum) {
  const int l = laneid(), n = l & 15, kb = (l < 16) ? 0 : 2, mb = (l < 16) ? 0 : 8;
  for (int r0 = 0; r0 < N_N; r0 += 16) {
    v8f acc = {};
    if (accum) {
#pragma unroll
      for (int r = 0; r < 8; ++r) acc[r] = D[(r0 + mb + r) * ldD + n];
    }
#pragma unroll
    for (int k0 = 0; k0 < P_N; k0 += 4) {
      v2f a, b;
      a.x = A[(r0 + n) * LDP + k0 + kb];      // row = r0 + (l&15)
      a.y = A[(r0 + n) * LDP + k0 + kb + 1];
      if (transB) {
        b.x = B16[n * LDP + k0 + kb];
        b.y = B16[n * LDP + k0 + kb + 1];
      } else {
        b.x = B16[(k0 + kb) * LDP + n];
        b.y = B16[(k0 + kb + 1) * LDP + n];
      }
      acc = wmma4(a, b, acc);
    }
#pragma unroll
    for (int r = 0; r < 8; ++r) D[(r0 + mb + r) * ldD + n] = acc[r];
  }
  __builtin_amdgcn_wave_barrier();
}

// D(16x16) = A16 * op(B16); all LDS stride LDP. Safe for D aliasing A (one wave,
// LDS ops in-order per wave; all loads precede stores).
__device__ __forceinline__ void mm_PxP(const float* A16, const float* B16, float* D,
                                       bool transB) {
  const int l = laneid(), n = l & 15, kb = (l < 16) ? 0 : 2, mb = (l < 16) ? 0 : 8;
  v8f acc = {};
#pragma unroll
  for (int k0 = 0; k0 < P_N; k0 += 4) {
    v2f a, b;
    a.x = A16[n * LDP + k0 + kb];
    a.y = A16[n * LDP + k0 + kb + 1];
    if (transB) {
      b.x = B16[n * LDP + k0 + kb];
      b.y = B16[n * LDP + k0 + kb + 1];
    } else {
      b.x = B16[(k0 + kb) * LDP + n];
      b.y = B16[(k0 + kb + 1) * LDP + n];
    }
    acc = wmma4(a, b, acc);
  }
#pragma unroll
  for (int r = 0; r < 8; ++r) D[(mb + r) * LDP + n] = acc[r];
  __builtin_amdgcn_wave_barrier();
}

// Z ~= inv(A) for 16x16 A (sing. values in (0,1]): Newton-Schulz, WMMA GEMMs.
__device__ __forceinline__ void ns_inverse(const float* A, float* Z, float* t1) {
  const int l = laneid();
  float fr = 0.f;
  for (int idx = l; idx < 256; idx += 32) {
    int r = idx >> 4, cc = idx & 15;
    float v = A[r * LDP + cc];
    fr += v * v;
  }
  fr = wave_sum(fr);
  const float alpha = 1.0f / fmaxf(fr, 1e-30f);
  for (int idx = l; idx < 256; idx += 32) {
    int r = idx >> 4, cc = idx & 15;
    Z[r * LDP + cc] = A[cc * LDP + r] * alpha;   // Z0 = A^T / ||A||_F^2
  }
  __builtin_amdgcn_wave_barrier();
  for (int it = 0; it < NS_ITERS; ++it) {
    mm_PxP(A, Z, t1, false);                     // t1 = A*Z
    for (int idx = l; idx < 256; idx += 32) {
      int r = idx >> 4, cc = idx & 15;
      t1[r * LDP + cc] = ((r == cc) ? 2.f : 0.f) - t1[r * LDP + cc];
    }
    __builtin_amdgcn_wave_barrier();
    mm_PxP(Z, t1, Z, false);                     // Z = Z*(2I - A*Z)
  }
}

// One-sided Jacobi: orthogonalize the 16 columns of T (nrows x 16, LDS).
// On exit: col k = u_k * s_k, V = right singular vectors (16x16).
__device__ __forceinline__ void jacobi_orth(float* T, float* V, int nrows) {
  const int l = laneid();
  for (int r = l; r < 16; r += 32) {
#pragma unroll
    for (int cc = 0; cc < 16; ++cc) V[r * LDP + cc] = (r == cc) ? 1.f : 0.f;
  }
  __builtin_amdgcn_wave_barrier();
  for (int sweep = 0; sweep < JACOBI_SWEEPS; ++sweep) {
    for (int i = 0; i < 15; ++i) {
      for (int j = i + 1; j < 16; ++j) {
        float app = 0.f, aqq = 0.f, apq = 0.f;
        for (int r = l; r < nrows; r += 32) {
          float x = T[r * LDP + i], y = T[r * LDP + j];
          app += x * x; aqq += y * y; apq += x * y;
        }
        app = wave_sum(app); aqq = wave_sum(aqq); apq = wave_sum(apq);
        if (fabsf(apq) > 1e-14f) {
          float tau = (aqq - app) / (2.f * apq);
          float t = copysignf(1.f, tau) / (fabsf(tau) + sqrtf(1.f + tau * tau));
          float cth = rsqrtf(1.f + t * t);
          float sth = t * cth;
          for (int r = l; r < nrows; r += 32) {
            float x = T[r * LDP + i], y = T[r * LDP + j];
            T[r * LDP + i] = cth * x - sth * y;
            T[r * LDP + j] = sth * x + cth * y;
          }
          for (int r = l; r < 16; r += 32) {
            float x = V[r * LDP + i], y = V[r * LDP + j];
            V[r * LDP + i] = cth * x - sth * y;
            V[r * LDP + j] = sth * x + cth * y;
          }
        }
        __builtin_amdgcn_wave_barrier();
      }
    }
  }
}

__device__ __forceinline__ float col_norm(const float* T, int k, int nrows) {
  float d = 0.f;
  for (int r = laneid(); r < nrows; r += 32) {
    float v = T[r * LDP + k];
    d += v * v;
  }
  return sqrtf(wave_sum(d));
}

// Per-wave LDS scratch.
struct WS {
  float *Xb, *T, *G;        // 96 x LDP
  float *P0, *P1, *P2, *P3; // 16 x LDP  (ytx, inv, tmp, V)
};

// Load one 96x16 tile (row-major, contiguous) from global into LDS (stride LDP).
__device__ __forceinline__ void wave_load_tile(const float* __restrict__ g, float* dst) {
  const int l = laneid();
#if RB_ASYNC_LDS
#pragma unroll
  for (int i = 0; i < 12; ++i) {
    int e = (i * 32 + l) * 4;        // 4 floats / lane / issue (b128)
    int r = e >> 4, cc = e & 15;     // row 16B-aligned in LDS (LDP*4 = 80)
    __builtin_amdgcn_global_load_async_to_lds_b128(
        (AS1 v4i*)(AS1 void*)(g + e),
        (AS3 v4i*)(AS3 void*)(dst + r * LDP + cc), 0, 0);
  }
  __builtin_amdgcn_s_wait_asynccnt(0);
#else
#pragma unroll
  for (int i = 0; i < 12; ++i) {
    int e = (i * 32 + l) * 4;
    int r = e >> 4, cc = e & 15;
    float4 v = *(const float4*)(g + e);
    *(float4*)(dst + r * LDP + cc) = v;
  }
#endif
  __builtin_amdgcn_wave_barrier();
}

// Grassmann log: tangent Log_{Ax}(Ay) -> w.G (96x16). Uses pinv via NS + Jacobi SVD.
__device__ __forceinline__ void wave_logmap(const float* Ax, const float* Ay, const WS& w) {
  const int l = laneid();
  mm_AtB96(Ay, Ax, w.P0);                        // ytx = Ay^T Ax
  ns_inverse(w.P0, w.P1, w.P2);                  // P1 ~= inv(ytx)
  mm_NxP_P(Ax, w.P0, w.G, LDP, true, false);     // G = Ax @ ytx^T
  for (int idx = l; idx < TILE_ELEMS; idx += 32) {
    int r = idx >> 4, cc = idx & 15;
    w.G[r * LDP + cc] = Ay[r * LDP + cc] - w.G[r * LDP + cc];
  }
  __builtin_amdgcn_wave_barrier();
  mm_NxP_P(w.G, w.P1, w.T, LDP, true, false);    // T = (Ay - G) @ inv^T
  jacobi_orth(w.T, w.P3, N_N);                   // T cols = u_k s_k, P3 = V
  for (int k = 0; k < 16; ++k) {
    float s = col_norm(w.T, k, N_N);
    float f = (s > 1e-12f) ? (atanf(s) / s) : 1.f;
    for (int r = l; r < N_N; r += 32) w.T[r * LDP + k] *= f;   // u_k * atan(s_k)
  }
  __builtin_amdgcn_wave_barrier();
  mm_NxP_P(w.T, w.P3, w.G, LDP, true, false);    // tangent = (U atan s) V^T
}

// Squared geodesic distance: sum of arccos(sigma(Ay^T Ax))^2.
__device__ __forceinline__ float wave_dist2(const float* Ax, const float* Ay, const WS& w) {
  mm_AtB96(Ay, Ax, w.P0);
  jacobi_orth(w.P0, w.P3, 16);
  float tot = 0.f;
  for (int k = 0; k < 16; ++k) {
    float s = fminf(col_norm(w.P0, k, 16), 1.f);
    float th = acosf(s);
    tot += th * th;
  }
  return tot;
}

// Grassmann exp: Exp_{Ax}(H) -> outp (row stride ldo). Destroys H, uses w.T.
__device__ __forceinline__ void wave_expmap(const float* Ax, float* H, float* outp,
                                            int ldo, const WS& w) {
  const int l = laneid();
  jacobi_orth(H, w.P3, N_N);                     // H cols = u_k s_k, P3 = V
  mm_NxP_P(Ax, w.P3, w.T, LDP, false, false);    // C1 = Ax @ V
  for (int k = 0; k < 16; ++k) {
    float s = col_norm(H, k, N_N);
    float cs = cosf(s);
    float f = (s > 1e-12f) ? (sinf(s) / s) : 1.f;
    for (int r = l; r < N_N; r += 32)
      w.T[r * LDP + k] = w.T[r * LDP + k] * cs + H[r * LDP + k] * f;
  }
  __builtin_amdgcn_wave_barrier();
  mm_NxP_P(w.T, w.P3, outp, ldo, true, false);   // out = comb @ V^T
}

__global__ __launch_bounds__(NW * 32) void rbngr_fused_kernel(
    const float* __restrict__ X, const float* __restrict__ bias,
    const float* __restrict__ shift, float* __restrict__ out) {
  __shared__ float sM0[N_N * LDP], sM[N_N * LDP], sQ[N_N * LDP], sW[N_N * LDP],
      sBias[N_N * LDP], sTan[N_N * LDP];
  __shared__ float wNP[NW][3][N_N * LDP];
  __shared__ float wPP[NW][4][16 * LDP];
  __shared__ float sVar, sSf;

  const int c = (int)blockIdx.x;
  const int wid = (int)(threadIdx.x >> 5);
  WS w{wNP[wid][0], wNP[wid][1], wNP[wid][2],
       wPP[wid][0], wPP[wid][1], wPP[wid][2], wPP[wid][3]};

  // Cooperative load of M0 = X[0,c], bias[c]; zero accumulators.
  for (int idx = (int)threadIdx.x; idx < TILE_ELEMS; idx += NW * 32) {
    int r = idx >> 4, cc = idx & 15;
    sM0[r * LDP + cc] = X[(size_t)c * TILE_ELEMS + idx];    // b = 0
    sBias[r * LDP + cc] = bias[(size_t)c * TILE_ELEMS + idx];
    sTan[r * LDP + cc] = 0.f;
  }
  if (threadIdx.x == 0) sVar = 0.f;
  __syncthreads();

  // ---- Phase 1: Karcher tangent mean (over b) at M0 -------------------------
  for (int b = wid; b < B_N; b += NW) {
    const float* gx = X + ((size_t)b * C_N + c) * TILE_ELEMS;
    if (b + NW < B_N)
      __builtin_prefetch(X + ((size_t)(b + NW) * C_N + c) * TILE_ELEMS + laneid() * 32, 0, 1);
    wave_load_tile(gx, w.Xb);
    wave_logmap(sM0, w.Xb, w);                   // tangent -> w.G
    for (int idx = laneid(); idx < TILE_ELEMS; idx += 32) {
      int r = idx >> 4, cc = idx & 15;
      atomicAdd(&sTan[r * LDP + cc], w.G[r * LDP + cc] * (1.f / B_N));
    }
  }
  __syncthreads();

  // M = Exp_{M0}(mean tangent)
  if (wid == 0) wave_expmap(sM0, sTan, sM, LDP, w);
  __syncthreads();

  // ---- Phase 1b: Frechet variance ------------------------------------------
  for (int b = wid; b < B_N; b += NW) {
    const float* gx = X + ((size_t)b * C_N + c) * TILE_ELEMS;
    wave_load_tile(gx, w.Xb);
    float d2 = wave_dist2(w.Xb, sM, w);
    if (laneid() == 0) atomicAdd(&sVar, d2 * (1.f / B_N));
  }
  __syncthreads();
  if (threadIdx.x == 0) sSf = shift[c] / sqrtf(sVar + 1e-5f);

  // ---- Phase 1c: transport frame. lfs = I + W Q^T ---------------------------
  if (wid == 0) {
    mm_AtB96(sBias, sM, w.P0);                   // ytx = bias^T M
    ns_inverse(w.P0, w.P1, w.P2);
    mm_NxP_P(sM, w.P0, w.G, LDP, true, false);   // G = M @ ytx^T
    for (int idx = laneid(); idx < TILE_ELEMS; idx += 32) {
      int r = idx >> 4, cc = idx & 15;
      w.G[r * LDP + cc] = sBias[r * LDP + cc] - w.G[r * LDP + cc];
    }
    __builtin_amdgcn_wave_barrier();
    mm_NxP_P(w.G, w.P1, w.T, LDP, true, false);  // T -> svd
    jacobi_orth(w.T, w.P3, N_N);                 // T cols = q_k s_k, P3 = R
    mm_NxP_P(sM, w.P3, w.G, LDP, false, false);  // MR = M @ R
    for (int k = 0; k < 16; ++k) {
      float s = col_norm(w.T, k, N_N);
      float st = atanf(s);
      float cs = cosf(st), sn = sinf(st);
      float qf = (s > 1e-12f) ? (1.f / s) : 0.f;
      for (int r = laneid(); r < N_N; r += 32) {
        float q = w.T[r * LDP + k] * qf;
        sQ[r * LDP + k] = q;
        sW[r * LDP + k] = q * (cs - 1.f) - w.G[r * LDP + k] * sn;
      }
    }
  }
  __syncthreads();

  // ---- Phase 2: normalize every sample --------------------------------------
  const float sf = sSf;
  for (int b = wid; b < B_N; b += NW) {
    const float* gx = X + ((size_t)b * C_N + c) * TILE_ELEMS;
    if (b + NW < B_N)
      __builtin_prefetch(X + ((size_t)(b + NW) * C_N + c) * TILE_ELEMS + laneid() * 32, 0, 1);
    wave_load_tile(gx, w.Xb);
    wave_logmap(sM, w.Xb, w);                    // delta -> w.G
    mm_AtB96(sQ, w.G, w.P0);                     // Hc = Q^T delta  (16x16)
    mm_NxP_P(sW, w.P0, w.G, LDP, false, true);   // delta += W @ Hc  (lfs@delta)
    for (int idx = laneid(); idx < TILE_ELEMS; idx += 32) {
      int r = idx >> 4, cc = idx & 15;
      w.G[r * LDP + cc] *= sf;
    }
    __builtin_amdgcn_wave_barrier();
    wave_expmap(sBias, w.G, out + ((size_t)b * C_N + c) * TILE_ELEMS, P_N, w);
  }
}

extern "C" void kernel_launch(void* const* d_in, const int* in_sizes, int n_in,
                              void* d_out, int out_size, void* d_ws, size_t ws_size,
                              hipStream_t stream) {
  (void)in_sizes; (void)n_in; (void)d_ws; (void)ws_size; (void)out_size;
  const float* X = (const float*)d_in[0];
  const float* bias = (const float*)d_in[1];
  const float* shift = (const float*)d_in[2];
  float* out = (float*)d_out;
  rbngr_fused_kernel<<<dim3(C_N), dim3(NW * 32), 0, stream>>>(X, bias, shift, out);
}